// NoHybridANFIS_38534446580294
// MI455X (gfx1250) — compile-verified
//
#include <hip/hip_runtime.h>
#include <hip/hip_bf16.h>

// Problem constants
constexpr int Bn = 1024;   // batch
constexpr int Dn = 32;     // dims
constexpr int Mn = 8;      // membership fns per dim
constexpr int Rn = 4096;   // rules
constexpr int Cn = 32;     // output channels
constexpr int Kk = 409;    // top-k = int(0.1 * R)

typedef float v2f __attribute__((ext_vector_type(2)));
typedef float v8f __attribute__((ext_vector_type(8)));

// ---------------------------------------------------------------------------
// K0: per-rule quadratic-form coefficients.
//   A[r,d] = -1/(2 w^2),  Bm[r,d] = c/w^2,  Cc[r] = sum_d -c^2/(2 w^2)
// One wave32 per rule, lane == d (D == 32 == warpSize).
// ---------------------------------------------------------------------------
__global__ void prep_rules_kernel(const float* __restrict__ centers, // D x M
                                  const float* __restrict__ widths,  // D x M
                                  const int*   __restrict__ rules,   // R x D
                                  float* __restrict__ Amat,          // R x D
                                  float* __restrict__ Bmat,          // R x D
                                  float* __restrict__ Cc)            // R
{
    const int wave = threadIdx.x >> 5;
    const int lane = threadIdx.x & 31;          // == d
    const int r    = blockIdx.x * (blockDim.x >> 5) + wave;
    const int m    = rules[r * Dn + lane];
    const float c  = centers[lane * Mn + m];
    const float w  = widths[lane * Mn + m];
    const float inv = 1.0f / (w * w);
    Amat[r * Dn + lane] = -0.5f * inv;
    Bmat[r * Dn + lane] =  c * inv;
    float cc = -0.5f * c * c * inv;
    #pragma unroll
    for (int off = 16; off >= 1; off >>= 1)
        cc += __shfl_down(cc, off, 32);
    if (lane == 0) Cc[r] = cc;
}

// ---------------------------------------------------------------------------
// K1: cons_sum[r,c] = sum_{j=0..D} consequents[r,j,c]
// ---------------------------------------------------------------------------
__global__ void cons_sum_kernel(const float* __restrict__ cons,  // R x (D+1) x C
                                float* __restrict__ csum)        // R x C
{
    const int idx = blockIdx.x * blockDim.x + threadIdx.x;   // < R*C
    const int r = idx / Cn;
    const int c = idx % Cn;
    float s = 0.0f;
    #pragma unroll 4
    for (int j = 0; j <= Dn; ++j)
        s += cons[((size_t)r * (Dn + 1) + j) * Cn + c];
    csum[idx] = s;
}

// ---------------------------------------------------------------------------
// K2: x2 = x*x ; xsum[b] = sum_d x[b,d] + 1. One wave per batch row.
// ---------------------------------------------------------------------------
__global__ void prep_x_kernel(const float* __restrict__ x,   // B x D
                              float* __restrict__ x2,        // B x D
                              float* __restrict__ xsum)      // B
{
    const int wave = threadIdx.x >> 5;
    const int lane = threadIdx.x & 31;
    const int b    = blockIdx.x * (blockDim.x >> 5) + wave;
    const float v  = x[b * Dn + lane];
    x2[b * Dn + lane] = v * v;
    float s = v;
    #pragma unroll
    for (int off = 16; off >= 1; off >>= 1)
        s += __shfl_down(s, off, 32);
    if (lane == 0) xsum[b] = s + 1.0f;
}

// ---------------------------------------------------------------------------
// K3: firing[b,r] = exp( X2 . A^T + X . Bm^T + Cc )  via V_WMMA_F32_16X16X4_F32
// One wave per 16x16 output tile; K loop over D in steps of 4.
// A-frag (16x4): lane&15 = M row; half-wave selects K,K+1 vs K+2,K+3.
// B-frag (4x16): lane&15 = N col; same half-wave K split (ISA 7.12.2).
// ---------------------------------------------------------------------------
__global__ void firing_wmma_kernel(const float* __restrict__ x,     // B x D
                                   const float* __restrict__ x2,    // B x D
                                   const float* __restrict__ Amat,  // R x D
                                   const float* __restrict__ Bmat,  // R x D
                                   const float* __restrict__ Cc,    // R
                                   float* __restrict__ firing)      // B x R
{
    const int wave  = threadIdx.x >> 5;
    const int lane  = threadIdx.x & 31;
    const int gw    = blockIdx.x * (blockDim.x >> 5) + wave;  // tile id
    const int rt    = gw % (Rn / 16);
    const int bt    = gw / (Rn / 16);
    const int b0    = bt * 16;
    const int r0    = rt * 16;
    const int mrow  = lane & 15;
    const int khalf = (lane >> 4) << 1;       // 0 or 2

    v8f acc = {};
    #pragma unroll
    for (int k0 = 0; k0 < Dn; k0 += 4) {
        v2f a, bb;
        // quadratic term: X2 (16x4) times Amat^T (4x16)
        a.x  = x2[(b0 + mrow) * Dn + k0 + khalf];
        a.y  = x2[(b0 + mrow) * Dn + k0 + khalf + 1];
        bb.x = Amat[(r0 + mrow) * Dn + k0 + khalf];
        bb.y = Amat[(r0 + mrow) * Dn + k0 + khalf + 1];
        acc = __builtin_amdgcn_wmma_f32_16x16x4_f32(false, a, false, bb,
                                                    (short)0, acc, false, false);
        // linear term: X (16x4) times Bmat^T (4x16)
        a.x  = x[(b0 + mrow) * Dn + k0 + khalf];
        a.y  = x[(b0 + mrow) * Dn + k0 + khalf + 1];
        bb.x = Bmat[(r0 + mrow) * Dn + k0 + khalf];
        bb.y = Bmat[(r0 + mrow) * Dn + k0 + khalf + 1];
        acc = __builtin_amdgcn_wmma_f32_16x16x4_f32(false, a, false, bb,
                                                    (short)0, acc, false, false);
    }

    const float cc  = Cc[r0 + mrow];          // constant term, per N column
    const int   bhi = (lane >> 4) * 8;        // C/D layout: hi half-wave = M+8
    #pragma unroll
    for (int i = 0; i < 8; ++i) {
        const float lf = acc[i] + cc;
        firing[(size_t)(b0 + i + bhi) * Rn + (r0 + mrow)] = __expf(lf);
    }
}

// ---------------------------------------------------------------------------
// K4: per-row top-K mask + normalization. One 256-thread block per batch row.
// MSB-first radix select over sign-flipped float keys; ties taken in index
// order (matches jax.lax.top_k); then norm_fs = firing*mask / (sum + eps).
// Overwrites the firing buffer (== norm_fs output region) in place.
// ---------------------------------------------------------------------------
__device__ __forceinline__ unsigned f2key(float f) {
    unsigned u = __float_as_uint(f);
    return (u & 0x80000000u) ? ~u : (u | 0x80000000u);
}

__global__ void topk_norm_kernel(float* __restrict__ norm,   // B x R (in: firing, out: norm_fs)
                                 float* __restrict__ maskout)// B x R
{
    constexpr int ELEMS = Rn / 256;              // 16 per thread
    const int b   = blockIdx.x;
    const int tid = threadIdx.x;
    float* row  = norm    + (size_t)b * Rn;
    float* mrow = maskout + (size_t)b * Rn;

    __shared__ float    vals[Rn];                // 16 KB
    __shared__ unsigned hist[256];
    __shared__ unsigned scan[256];
    __shared__ float    ssum[256];
    __shared__ unsigned sbyte, scum;

    #pragma unroll
    for (int j = 0; j < ELEMS; ++j)
        vals[tid * ELEMS + j] = row[tid * ELEMS + j];
    __syncthreads();

    // ---- radix select: find key of the K-th largest element ----
    unsigned prefix = 0;
    int remaining = Kk;
    for (int shift = 24; shift >= 0; shift -= 8) {
        hist[tid] = 0;
        __syncthreads();
        const unsigned hm = (shift == 24) ? 0u : (0xFFFFFFFFu << (shift + 8));
        #pragma unroll
        for (int j = 0; j < ELEMS; ++j) {
            const unsigned u = f2key(vals[tid * ELEMS + j]);
            if ((u & hm) == (prefix & hm))
                atomicAdd(&hist[(u >> shift) & 0xFFu], 1u);
        }
        __syncthreads();
        if (tid == 0) {
            unsigned cum = 0; int sel = 0;
            for (int v = 255; v >= 0; --v) {
                const unsigned h = hist[v];
                if (cum + h >= (unsigned)remaining) { sel = v; break; }
                cum += h;
            }
            sbyte = (unsigned)sel; scum = cum;
        }
        __syncthreads();
        prefix   |= (sbyte << shift);
        remaining -= (int)scum;
        __syncthreads();
    }
    // key > prefix: always selected (count == Kk - remaining).
    // key == prefix: select first `remaining` in index order.

    int myties = 0;
    #pragma unroll
    for (int j = 0; j < ELEMS; ++j)
        if (f2key(vals[tid * ELEMS + j]) == prefix) ++myties;
    scan[tid] = (unsigned)myties;
    __syncthreads();
    for (int off = 1; off < 256; off <<= 1) {     // inclusive block scan
        unsigned v = (tid >= off) ? scan[tid - off] : 0u;
        __syncthreads();
        scan[tid] += v;
        __syncthreads();
    }
    const unsigned tiebase = scan[tid] - (unsigned)myties;  // exclusive prefix

    float lsum = 0.0f;
    int trank = 0;
    #pragma unroll
    for (int j = 0; j < ELEMS; ++j) {
        const float v = vals[tid * ELEMS + j];
        const unsigned u = f2key(v);
        float m = 0.0f;
        if (u > prefix) m = 1.0f;
        else if (u == prefix) {
            if ((int)(tiebase + (unsigned)trank) < remaining) m = 1.0f;
            ++trank;
        }
        mrow[tid * ELEMS + j] = m;
        const float fv = v * m;
        vals[tid * ELEMS + j] = fv;
        lsum += fv;
    }
    ssum[tid] = lsum;
    __syncthreads();
    for (int off = 128; off > 0; off >>= 1) {
        if (tid < off) ssum[tid] += ssum[tid + off];
        __syncthreads();
    }
    const float inv = 1.0f / (ssum[0] + 1e-9f);
    #pragma unroll
    for (int j = 0; j < ELEMS; ++j)
        row[tid * ELEMS + j] = vals[tid * ELEMS + j] * inv;
}

// ---------------------------------------------------------------------------
// K5: out[b,c] = xsum[b] * (norm_fs @ cons_sum)  via V_WMMA_F32_16X16X4_F32.
// One wave per 16x16 tile, K loop over R=4096.
// ---------------------------------------------------------------------------
__global__ void output_wmma_kernel(const float* __restrict__ norm,  // B x R
                                   const float* __restrict__ csum,  // R x C
                                   const float* __restrict__ xsum,  // B
                                   float* __restrict__ out)         // B x C
{
    const int wave  = threadIdx.x >> 5;
    const int lane  = threadIdx.x & 31;
    const int gw    = blockIdx.x * (blockDim.x >> 5) + wave;  // 0..127
    const int ct    = gw & ((Cn / 16) - 1);
    const int bt    = gw / (Cn / 16);
    const int b0    = bt * 16;
    const int c0    = ct * 16;
    const int mrow  = lane & 15;
    const int khalf = (lane >> 4) << 1;

    v8f acc = {};
    for (int k0 = 0; k0 < Rn; k0 += 4) {
        v2f a, bb;
        a.x  = norm[(size_t)(b0 + mrow) * Rn + k0 + khalf];
        a.y  = norm[(size_t)(b0 + mrow) * Rn + k0 + khalf + 1];
        bb.x = csum[(size_t)(k0 + khalf) * Cn + c0 + mrow];
        bb.y = csum[(size_t)(k0 + khalf + 1) * Cn + c0 + mrow];
        acc = __builtin_amdgcn_wmma_f32_16x16x4_f32(false, a, false, bb,
                                                    (short)0, acc, false, false);
    }
    const int bhi = (lane >> 4) * 8;
    #pragma unroll
    for (int i = 0; i < 8; ++i) {
        const int brow = b0 + i + bhi;
        out[(size_t)brow * Cn + c0 + mrow] = acc[i] * xsum[brow];
    }
}

// ---------------------------------------------------------------------------
// Launch
// ---------------------------------------------------------------------------
extern "C" void kernel_launch(void* const* d_in, const int* in_sizes, int n_in,
                              void* d_out, int out_size, void* d_ws, size_t ws_size,
                              hipStream_t stream) {
    const float* x       = (const float*)d_in[0];   // B x D
    const float* centers = (const float*)d_in[1];   // D x M
    const float* widths  = (const float*)d_in[2];   // D x M
    const float* cons    = (const float*)d_in[3];   // R x (D+1) x C
    const int*   rules   = (const int*)d_in[4];     // R x D

    float* out_rule = (float*)d_out;                         // B x C
    float* out_norm = out_rule + (size_t)Bn * Cn;            // B x R (also firing scratch)
    float* out_mask = out_norm + (size_t)Bn * Rn;            // B x R

    float* ws   = (float*)d_ws;
    float* Amat = ws;                       // R*D
    float* Bmat = Amat + (size_t)Rn * Dn;   // R*D
    float* Cc   = Bmat + (size_t)Rn * Dn;   // R
    float* csum = Cc + Rn;                  // R*C
    float* x2   = csum + (size_t)Rn * Cn;   // B*D
    float* xsum = x2 + (size_t)Bn * Dn;     // B
    // total ws use: ~431K floats (~1.7 MB)

    prep_rules_kernel<<<Rn / 8, 256, 0, stream>>>(centers, widths, rules, Amat, Bmat, Cc);
    cons_sum_kernel<<<(Rn * Cn) / 256, 256, 0, stream>>>(cons, csum);
    prep_x_kernel<<<Bn / 8, 256, 0, stream>>>(x, x2, xsum);

    const int firing_tiles = (Bn / 16) * (Rn / 16);          // 16384 waves
    firing_wmma_kernel<<<firing_tiles / 8, 256, 0, stream>>>(x, x2, Amat, Bmat, Cc, out_norm);

    topk_norm_kernel<<<Bn, 256, 0, stream>>>(out_norm, out_mask);

    const int out_tiles = (Bn / 16) * (Cn / 16);             // 128 waves
    output_wmma_kernel<<<out_tiles / 4, 128, 0, stream>>>(out_norm, csum, xsum, out_rule);
}